// ReduceState_39891656245702
// MI455X (gfx1250) — compile-verified
//
#include <hip/hip_runtime.h>
#include <hip/hip_bf16.h>
#include <stdint.h>

// Scatter-mean: out[s,:] = mean(features[i,:] for index[i]==s), D=64, f32.
// Counting-sort rows by segment, then coalesced gather-reduce.
// Roofline: ~353 MB total traffic -> ~15us @ 23.3 TB/s. Avoids 80M global
// f32 atomics (L2 atomic-rate bound, ~5-10x slower than streaming).

#define D_FEAT 64
#define SCAN_THREADS 1024
#define SCAN_TILE 8192                   // ints per tile -> 32KB LDS stage
#define SCAN_PER_T (SCAN_TILE / SCAN_THREADS)  // 8 ints per thread

// ---------------- Kernel 1: zero segment counts ----------------
__global__ void k_zero_counts(int* __restrict__ counts, int S) {
    int i = blockIdx.x * blockDim.x + threadIdx.x;
    if (i < S) counts[i] = 0;
}

// ---------------- Kernel 2: histogram of segment ids ----------------
__global__ void k_hist(const int* __restrict__ idx, int* __restrict__ counts, int n) {
    int i = blockIdx.x * blockDim.x + threadIdx.x;
    int stride = gridDim.x * blockDim.x;
    for (; i < n; i += stride) {
        if (i + stride < n) {
            __builtin_prefetch(&idx[i + stride], 0, 0);  // global_prefetch_b8
        }
        atomicAdd(&counts[idx[i]], 1);  // non-returning int atomic (STOREcnt path)
    }
}

// ---------------- Kernel 3: tiled exclusive scan (single workgroup) ----------
// Each 32KB tile of counts is DMA-staged into LDS with the gfx1250 async
// global->LDS path (no VGPR return traffic, perfectly coalesced), then scanned
// from LDS. Carry propagates across tiles.
__global__ void k_scan(const int* __restrict__ counts, int* __restrict__ offsets,
                       int* __restrict__ cursor, int S) {
    __shared__ __align__(16) int tile[SCAN_TILE];
    __shared__ int partials[SCAN_THREADS];
    __shared__ int carry_sh;
    const int t = threadIdx.x;
    if (t == 0) carry_sh = 0;
    __syncthreads();

    const int ntiles = (S + SCAN_TILE - 1) / SCAN_TILE;
    for (int ti = 0; ti < ntiles; ++ti) {
        const int tbase = ti * SCAN_TILE;

        // ---- async DMA stage: 8 ints/thread = two b128 transfers ----
        {
            uint64_t g0   = (uint64_t)(uintptr_t)(counts + tbase + t * SCAN_PER_T);
            uint64_t g1   = g0 + 16;
            uint32_t lds0 = (uint32_t)(uintptr_t)(&tile[t * SCAN_PER_T]);
            uint32_t lds1 = lds0 + 16;
            asm volatile("global_load_async_to_lds_b128 %0, %1, off"
                         :: "v"(lds0), "v"(g0) : "memory");
            asm volatile("global_load_async_to_lds_b128 %0, %1, off"
                         :: "v"(lds1), "v"(g1) : "memory");
            asm volatile("s_wait_asynccnt 0x0" ::: "memory");
        }
        __syncthreads();

        // ---- per-thread local sums from LDS ----
        int vals[SCAN_PER_T];
        int local = 0;
#pragma unroll
        for (int k = 0; k < SCAN_PER_T; ++k) {
            int gi = tbase + t * SCAN_PER_T + k;
            int c = (gi < S) ? tile[t * SCAN_PER_T + k] : 0;
            vals[k] = c;
            local += c;
        }
        partials[t] = local;
        __syncthreads();

        // ---- Hillis-Steele inclusive scan over 1024 partials ----
        for (int off = 1; off < SCAN_THREADS; off <<= 1) {
            int v = (t >= off) ? partials[t - off] : 0;
            __syncthreads();
            partials[t] += v;
            __syncthreads();
        }

        // ---- write exclusive offsets + cursors for this tile ----
        int run = carry_sh + ((t == 0) ? 0 : partials[t - 1]);
#pragma unroll
        for (int k = 0; k < SCAN_PER_T; ++k) {
            int gi = tbase + t * SCAN_PER_T + k;
            if (gi < S) { offsets[gi] = run; cursor[gi] = run; }
            run += vals[k];
        }
        __syncthreads();
        if (t == SCAN_THREADS - 1) carry_sh += partials[t];  // tile total
        __syncthreads();
    }
    if (t == SCAN_THREADS - 1) offsets[S] = carry_sh;  // == n
}

// ---------------- Kernel 4: scatter row ids into buckets ----------------
__global__ void k_scatter(const int* __restrict__ idx, int* __restrict__ cursor,
                          int* __restrict__ rowids, int n) {
    int i = blockIdx.x * blockDim.x + threadIdx.x;
    int stride = gridDim.x * blockDim.x;
    for (; i < n; i += stride) {
        if (i + stride < n) __builtin_prefetch(&idx[i + stride], 0, 0);
        int s = idx[i];
        int pos = atomicAdd(&cursor[s], 1);  // returning int atomic (1.25M total)
        rowids[pos] = i;
    }
}

// ---------------- Kernel 5: gather-reduce, one 16-lane half-wave/segment -----
// wave32: 2 segments per wave. Lanes cooperatively load 16 rowids with ONE
// coalesced 64B transfer, then broadcast via lane-shuffle so all 16 feature-row
// b128 gathers are address-independent and can be kept in flight together.
__global__ void k_reduce(const float* __restrict__ feat,
                         const int* __restrict__ offsets,
                         const int* __restrict__ rowids,
                         float* __restrict__ out, int S) {
    const int lane = threadIdx.x & 15;
    const int grp  = threadIdx.x >> 4;
    const int s    = blockIdx.x * 16 + grp;
    if (s >= S) return;

    const int start = offsets[s];
    const int end   = offsets[s + 1];

    float4 acc = make_float4(0.f, 0.f, 0.f, 0.f);
    for (int base = start; base < end; base += 16) {
        int m = end - base;
        if (m > 16) m = 16;
        // coalesced chunk load of up to 16 row ids (one 64B transfer per group)
        int myrid = (lane < m) ? rowids[base + lane] : 0;
        if (base + 16 < end) __builtin_prefetch(&rowids[base + 16 + lane], 0, 0);
#pragma unroll
        for (int k = 0; k < 16; ++k) {
            if (k >= m) break;
            int row = __shfl(myrid, k, 16);  // wave32 lane-permute broadcast
            const float4 v =
                *reinterpret_cast<const float4*>(feat + (size_t)row * D_FEAT + lane * 4);
            acc.x += v.x; acc.y += v.y; acc.z += v.z; acc.w += v.w;
        }
    }

    const float denom = (float)((end - start) > 1 ? (end - start) : 1);
    acc.x /= denom; acc.y /= denom; acc.z /= denom; acc.w /= denom;
    *reinterpret_cast<float4*>(out + (size_t)s * D_FEAT + lane * 4) = acc;
}

extern "C" void kernel_launch(void* const* d_in, const int* in_sizes, int n_in,
                              void* d_out, int out_size, void* d_ws, size_t ws_size,
                              hipStream_t stream) {
    const float* feat = (const float*)d_in[0];
    const int*   idx  = (const int*)d_in[1];   // JAX default x64-off -> int32
    float*       out  = (float*)d_out;

    const int n = in_sizes[1];          // rows (1.25M)
    const int S = out_size / D_FEAT;    // segments (50K)

    // Workspace: counts[S], offsets[S+1], cursor[S], rowids[n]  (~5.6 MB).
    // NOTE: k_scan's async stage may read up to 224KB past counts[] start
    // (tile overshoot) -- that region is inside d_ws, values are masked out.
    char* ws = (char*)d_ws;
    size_t off = 0;
    int* counts  = (int*)(ws + off); off += ((size_t)S * 4 + 255) & ~(size_t)255;
    int* offsets = (int*)(ws + off); off += ((size_t)(S + 1) * 4 + 255) & ~(size_t)255;
    int* cursor  = (int*)(ws + off); off += ((size_t)S * 4 + 255) & ~(size_t)255;
    int* rowids  = (int*)(ws + off);
    (void)ws_size; (void)n_in;

    const int T = 256;

    k_zero_counts<<<(S + T - 1) / T, T, 0, stream>>>(counts, S);
    k_hist<<<2048, T, 0, stream>>>(idx, counts, n);
    k_scan<<<1, SCAN_THREADS, 0, stream>>>(counts, offsets, cursor, S);
    k_scatter<<<2048, T, 0, stream>>>(idx, cursor, rowids, n);
    k_reduce<<<(S + 15) / 16, T, 0, stream>>>(feat, offsets, rowids, out, S);
}